// SetToSequenceGroup_decoder_21672404976117
// MI455X (gfx1250) — compile-verified
//
#include <hip/hip_runtime.h>
#include <math.h>

#define HID 1024
#define NMEM 8192
#define GEMV_K 1024          // fixed inner dimension for every mat-vec here
#define GEMV_KLEN (GEMV_K/4) // split-K across 4 waves -> 256 per wave

typedef __attribute__((ext_vector_type(2))) float v2f;
typedef __attribute__((ext_vector_type(8))) float v8f;

// ---------------------------------------------------------------------------
// 16-row-tile mat-vec using V_WMMA_F32_16X16X4_F32 (K fixed at 1024).
//   y[row] = sum_k W[row,k] * x[k]  (+ bias)
// W row-major [rows, 1024]. One workgroup = 4 waves, split-K by 4, LDS reduce.
// A tile: lane (m = lane&15) holds row m; lanes 0-15 carry K-pair {k,k+1},
// lanes 16-31 carry {k+2,k+3} (ISA 16x4 f32 A layout). B: vector chunk in
// column 0 only (lanes 0 and 16), other columns zeroed via multiply so EXEC
// stays all-ones (WMMA requirement). D column 0 = 16 dot-product partials.
// Constant trip count -> scalar (SALU) loop control; 4 independent WMMA
// accumulator chains; 8 b64 loads per iteration issued as clauses.
// bias_mode: 0 = none, 1 = per-row vector, 2 = scalar bias[0].
// ---------------------------------------------------------------------------
__global__ void __launch_bounds__(128) gemv16_wmma(
    const float* __restrict__ W, const float* __restrict__ x,
    const float* __restrict__ bias, int bias_mode,
    float* __restrict__ y)
{
    __shared__ float part[4][16];
    const int tid  = threadIdx.x;
    const int wave = tid >> 5;
    const int lane = tid & 31;
    const int m    = lane & 15;
    const int koff = (lane >> 4) << 1;      // 0 (lanes 0-15) or 2 (lanes 16-31)
    const int row0 = blockIdx.x << 4;

    const int k0 = wave * GEMV_KLEN;

    const float* wp = W + (size_t)(row0 + m) * (size_t)GEMV_K + (size_t)(k0 + koff);
    const float* xp = x + k0 + koff;
    const float  zm = (m == 0) ? 1.0f : 0.0f;   // keep only B column 0

    v8f c0 = {}, c1 = {}, c2 = {}, c3 = {};

    // 16 K-elements per iteration, constant trip count (16 iterations)
    for (int it = 0; it < GEMV_KLEN / 16; ++it) {
        v2f a0 = *(const v2f*)(wp + 0);
        v2f a1 = *(const v2f*)(wp + 4);
        v2f a2 = *(const v2f*)(wp + 8);
        v2f a3 = *(const v2f*)(wp + 12);
        v2f b0 = *(const v2f*)(xp + 0);
        v2f b1 = *(const v2f*)(xp + 4);
        v2f b2 = *(const v2f*)(xp + 8);
        v2f b3 = *(const v2f*)(xp + 12);
        b0.x *= zm; b0.y *= zm;
        b1.x *= zm; b1.y *= zm;
        b2.x *= zm; b2.y *= zm;
        b3.x *= zm; b3.y *= zm;
        // (neg_a, A, neg_b, B, c_mod, C, reuse_a, reuse_b)
        c0 = __builtin_amdgcn_wmma_f32_16x16x4_f32(false, a0, false, b0,
                                                   (short)0, c0, false, false);
        c1 = __builtin_amdgcn_wmma_f32_16x16x4_f32(false, a1, false, b1,
                                                   (short)0, c1, false, false);
        c2 = __builtin_amdgcn_wmma_f32_16x16x4_f32(false, a2, false, b2,
                                                   (short)0, c2, false, false);
        c3 = __builtin_amdgcn_wmma_f32_16x16x4_f32(false, a3, false, b3,
                                                   (short)0, c3, false, false);
        wp += 16;
        xp += 16;
    }
    c0 += c1;
    c2 += c3;
    c0 += c2;

    // D column 0: lane 0 holds rows 0..7 (acc VGPRs 0..7), lane 16 rows 8..15
    if (lane == 0) {
#pragma unroll
        for (int i = 0; i < 8; ++i) part[wave][i] = c0[i];
    } else if (lane == 16) {
#pragma unroll
        for (int i = 0; i < 8; ++i) part[wave][8 + i] = c0[i];
    }
    __syncthreads();
    if (tid < 16) {
        float s = part[0][tid] + part[1][tid] + part[2][tid] + part[3][tid];
        if (bias_mode == 1)      s += bias[row0 + tid];
        else if (bias_mode == 2) s += bias[0];
        y[row0 + tid] = s;
    }
}

// ---------------------------------------------------------------------------
// GRU gate fusion: h_new = (1-z)*n + z*h  (PyTorch GRUCell math)
// ---------------------------------------------------------------------------
__global__ void gru_combine(const float* __restrict__ gi, const float* __restrict__ gh,
                            const float* __restrict__ h,
                            float* __restrict__ hn_ws, float* __restrict__ hn_out)
{
    int j = blockIdx.x * blockDim.x + threadIdx.x;
    if (j >= HID) return;
    float r = 1.0f / (1.0f + expf(-(gi[j]           + gh[j])));
    float z = 1.0f / (1.0f + expf(-(gi[HID + j]     + gh[HID + j])));
    float n = tanhf(gi[2 * HID + j] + r * gh[2 * HID + j]);
    float v = (1.0f - z) * n + z * h[j];
    hn_ws[j]  = v;
    hn_out[j] = v;
}

// ---------------------------------------------------------------------------
// mask pass-through
// ---------------------------------------------------------------------------
__global__ void copy_mask(const float* __restrict__ mask, float* __restrict__ out, int n)
{
    int i = blockIdx.x * blockDim.x + threadIdx.x;
    if (i < n) out[i] = mask[i];
}

// ---------------------------------------------------------------------------
// Single-workgroup log_softmax + argmax over e[N]. Writes:
//   out[0]            = (float)argmax
//   out[attn_off + i] = e[i] - (max + log(sum exp(e - max)))
// Tie-break: lowest index (matches jnp.argmax).
// ---------------------------------------------------------------------------
__global__ void __launch_bounds__(1024) logsoftmax_argmax(
    const float* __restrict__ e, float* __restrict__ out, int attn_off, int n)
{
    __shared__ float smax[1024];
    __shared__ int   sidx[1024];
    __shared__ float ssum[1024];
    const int t = threadIdx.x;

    float m = -INFINITY; int mi = n;
    for (int i = t; i < n; i += 1024) {
        float v = e[i];
        if (v > m) { m = v; mi = i; }
    }
    smax[t] = m; sidx[t] = mi;
    __syncthreads();
    for (int s = 512; s > 0; s >>= 1) {
        if (t < s) {
            float a = smax[t], b = smax[t + s];
            if (b > a || (b == a && sidx[t + s] < sidx[t])) {
                smax[t] = b; sidx[t] = sidx[t + s];
            }
        }
        __syncthreads();
    }
    const float gmax = smax[0];
    const int   gidx = sidx[0];
    __syncthreads();

    float s = 0.0f;
    for (int i = t; i < n; i += 1024) s += expf(e[i] - gmax);
    ssum[t] = s;
    __syncthreads();
    for (int st = 512; st > 0; st >>= 1) {
        if (t < st) ssum[t] += ssum[t + st];
        __syncthreads();
    }
    const float lse = gmax + logf(ssum[0]);

    for (int i = t; i < n; i += 1024) out[attn_off + i] = e[i] - lse;
    if (t == 0) out[0] = (float)gidx;
}

// ---------------------------------------------------------------------------
// Host launcher
// ---------------------------------------------------------------------------
extern "C" void kernel_launch(void* const* d_in, const int* in_sizes, int n_in,
                              void* d_out, int out_size, void* d_ws, size_t ws_size,
                              hipStream_t stream)
{
    (void)in_sizes; (void)n_in; (void)out_size; (void)ws_size;

    const float* input = (const float*)d_in[0];   // [1, H]
    const float* h_t   = (const float*)d_in[1];   // [1, H]
    const float* mem   = (const float*)d_in[2];   // [N, H]
    const float* mask  = (const float*)d_in[3];   // [N]
    const float* w_ih  = (const float*)d_in[4];   // [3H, H]
    const float* w_hh  = (const float*)d_in[5];   // [3H, H]
    const float* b_ih  = (const float*)d_in[6];   // [3H]
    const float* b_hh  = (const float*)d_in[7];   // [3H]
    const float* w_bil = (const float*)d_in[8];   // [H, H]
    const float* b_bil = (const float*)d_in[9];   // scalar

    float* out = (float*)d_out;                   // [1 + H + N + N]
    float* ws  = (float*)d_ws;
    float* gi  = ws;                              // 3H
    float* gh  = ws + 3 * HID;                    // 3H
    float* hn  = ws + 6 * HID;                    // H
    float* u   = ws + 7 * HID;                    // H
    float* e   = ws + 8 * HID;                    // N

    const int attn_off = 1 + HID + NMEM;

    // GRU input/hidden projections: [3H,H] @ [H] (+bias), 192 tiles each
    gemv16_wmma<<<3 * HID / 16, 128, 0, stream>>>(w_ih, input, b_ih, 1, gi);
    gemv16_wmma<<<3 * HID / 16, 128, 0, stream>>>(w_hh, h_t,   b_hh, 1, gh);

    // Gate fusion -> h_new (also written to output slot)
    gru_combine<<<HID / 256, 256, 0, stream>>>(gi, gh, h_t, hn, out + 1);

    // u = W_bil @ h_new
    gemv16_wmma<<<HID / 16, 128, 0, stream>>>(w_bil, hn, nullptr, 0, u);

    // e = mem_block @ u + b_bil  (512 tiles x 4 waves = 2048 waves in flight)
    gemv16_wmma<<<NMEM / 16, 128, 0, stream>>>(mem, u, b_bil, 2, e);

    // mask pass-through
    copy_mask<<<NMEM / 256, 256, 0, stream>>>(mask, out + 1 + HID, NMEM);

    // log_softmax + argmax -> attn, target
    logsoftmax_argmax<<<1, 1024, 0, stream>>>(e, out, attn_off, NMEM);
}